// StructuralEncoder_84928683311673
// MI455X (gfx1250) — compile-verified
//
#include <hip/hip_runtime.h>
#include <hip/hip_bf16.h>

// ---------------------------------------------------------------------------
// GAT x2 for MI455X (gfx1250, wave32).
//   layer GEMM  : V_WMMA_F32_16X16X4_F32 (fp32 matrix pipe, exact precision)
//   edge passes : ordered-uint atomicMax (segment max), global f32 atomic adds
//   all elementwise / gather paths vectorized to b128 where possible
// ---------------------------------------------------------------------------

typedef float v2f __attribute__((ext_vector_type(2)));
typedef float v8f __attribute__((ext_vector_type(8)));

#define D128 128
#define HEADS 8
#define CPH 16
#define NEG_SLOPE 0.2f

// ---- order-preserving float<->uint encoding for atomicMax on floats --------
__device__ __forceinline__ unsigned ordered_enc(float f) {
  unsigned u = __float_as_uint(f);
  return (u & 0x80000000u) ? ~u : (u | 0x80000000u);
}
__device__ __forceinline__ float ordered_dec(unsigned u) {
  return (u & 0x80000000u) ? __uint_as_float(u & 0x7fffffffu)
                           : __uint_as_float(~u);
}
__device__ __forceinline__ float lrelu(float v) {
  return v > 0.f ? v : NEG_SLOPE * v;
}

// ---------------------------------------------------------------------------
// GEMM: Hout[N,128] = A[N,128] * B[128,128], one wave per 16x16 output tile.
// V_WMMA_F32_16X16X4_F32, 32 MACs along K=128.
// A frag (16x4 f32, 2 VGPR): lanes 0-15 -> (M=lane, K={0,1}), lanes 16-31 ->
// (M=lane-16, K={2,3}).  B frag (4x16): row K striped across lanes per VGPR.
// C/D (8 VGPR): VGPR r: lanes 0-15 -> (M=r, N=lane); lanes 16-31 -> (M=8+r).
// ---------------------------------------------------------------------------
__global__ __launch_bounds__(64)
void gemm128_wmma_f32(const float* __restrict__ A,
                      const float* __restrict__ B,
                      float* __restrict__ Hout,
                      int Nrows) {
  const int wave = blockIdx.x * 2 + (threadIdx.x >> 5);
  const int lane = threadIdx.x & 31;
  const int tilesM = (Nrows + 15) >> 4;
  const int tm = wave >> 3;       // 8 tiles along N (128/16)
  const int tn = wave & 7;
  if (tm >= tilesM) return;

  const int mLane = lane & 15;
  const int kHalf = (lane >> 4) << 1;     // 0 or 2
  const int col   = tn * 16 + mLane;

  const float* __restrict__ Arow = A + (size_t)(tm * 16 + mLane) * D128;

  v8f acc = {};
#pragma unroll
  for (int k = 0; k < D128; k += 4) {
    v2f a = *(const v2f*)(Arow + k + kHalf);
    v2f b;
    b.x = B[(size_t)(k + kHalf + 0) * D128 + col];
    b.y = B[(size_t)(k + kHalf + 1) * D128 + col];
    acc = __builtin_amdgcn_wmma_f32_16x16x4_f32(
        /*neg_a=*/false, a, /*neg_b=*/false, b,
        /*c_mod=*/(short)0, acc, /*reuse_a=*/false, /*reuse_b=*/false);
  }

  const int cmBase = tm * 16 + ((lane >> 4) << 3);
#pragma unroll
  for (int r = 0; r < 8; ++r)
    Hout[(size_t)(cmBase + r) * D128 + col] = acc[r];
}

// ---------------------------------------------------------------------------
// Per-node attention logits: asrc[n,h] = dot(h[n,h,:], a_src[h,:]), same adst.
// b128 loads: 4x float4 per head slice.
// ---------------------------------------------------------------------------
__global__ __launch_bounds__(256)
void node_alpha(const float* __restrict__ h,
                const float* __restrict__ a_src,
                const float* __restrict__ a_dst,
                float* __restrict__ asrc, float* __restrict__ adst, int Nn) {
  int i = blockIdx.x * blockDim.x + threadIdx.x;     // i = n*8 + head
  if (i >= Nn * HEADS) return;
  const int n = i >> 3, hh = i & 7;
  const float4* hp = (const float4*)(h + (size_t)n * D128 + hh * CPH);
  const float4* as = (const float4*)(a_src + hh * CPH);
  const float4* ad = (const float4*)(a_dst + hh * CPH);
  float s0 = 0.f, s1 = 0.f;
#pragma unroll
  for (int q = 0; q < 4; ++q) {
    float4 v = hp[q], a = as[q], d = ad[q];
    s0 += v.x * a.x + v.y * a.y + v.z * a.z + v.w * a.w;
    s1 += v.x * d.x + v.y * d.y + v.z * d.z + v.w * d.w;
  }
  asrc[i] = s0;
  adst[i] = s1;
}

// ---------------------------------------------------------------------------
// Zero the accumulator / denom, reset encoded max. float4-vectorized.
// ---------------------------------------------------------------------------
__global__ __launch_bounds__(256)
void init_bufs(float4* __restrict__ acc4, float* __restrict__ denom,
               unsigned* __restrict__ emax, int Nn) {
  int i = blockIdx.x * blockDim.x + threadIdx.x;
  if (i < Nn * (D128 / 4)) acc4[i] = make_float4(0.f, 0.f, 0.f, 0.f);
  if (i < Nn * HEADS) { denom[i] = 0.f; emax[i] = 0u; }
}

// ---------------------------------------------------------------------------
// Edge pass 1: segment max of leaky_relu(asrc[row]+adst[col]) into emax[col].
// One thread per edge; self-loops are edges [E, E+N). Logits via 2x b128.
// ---------------------------------------------------------------------------
__global__ __launch_bounds__(256)
void edge_pass_max(const int* __restrict__ ei, int E, int Nn,
                   const float* __restrict__ asrc,
                   const float* __restrict__ adst,
                   unsigned* __restrict__ emax) {
  int e = blockIdx.x * blockDim.x + threadIdx.x;
  const int EE = E + Nn;
  if (e >= EE) return;
  int r, c;
  if (e < E) { r = ei[e]; c = ei[E + e]; } else { r = c = e - E; }
  float sa[8], sd[8];
  *(float4*)&sa[0] = ((const float4*)(asrc + r * HEADS))[0];
  *(float4*)&sa[4] = ((const float4*)(asrc + r * HEADS))[1];
  *(float4*)&sd[0] = ((const float4*)(adst + c * HEADS))[0];
  *(float4*)&sd[4] = ((const float4*)(adst + c * HEADS))[1];
#pragma unroll
  for (int hh = 0; hh < HEADS; ++hh)
    atomicMax(&emax[c * HEADS + hh], ordered_enc(lrelu(sa[hh] + sd[hh])));
}

// ---------------------------------------------------------------------------
// Edge pass 2: denom[col] += exp(e - emax[col]).
// ---------------------------------------------------------------------------
__global__ __launch_bounds__(256)
void edge_pass_denom(const int* __restrict__ ei, int E, int Nn,
                     const float* __restrict__ asrc,
                     const float* __restrict__ adst,
                     const unsigned* __restrict__ emax,
                     float* __restrict__ denom) {
  int e = blockIdx.x * blockDim.x + threadIdx.x;
  const int EE = E + Nn;
  if (e >= EE) return;
  int r, c;
  if (e < E) { r = ei[e]; c = ei[E + e]; } else { r = c = e - E; }
  float sa[8], sd[8];
  *(float4*)&sa[0] = ((const float4*)(asrc + r * HEADS))[0];
  *(float4*)&sa[4] = ((const float4*)(asrc + r * HEADS))[1];
  *(float4*)&sd[0] = ((const float4*)(adst + c * HEADS))[0];
  *(float4*)&sd[4] = ((const float4*)(adst + c * HEADS))[1];
#pragma unroll
  for (int hh = 0; hh < HEADS; ++hh) {
    float v = lrelu(sa[hh] + sd[hh]);
    float m = ordered_dec(emax[c * HEADS + hh]);
    atomicAdd(&denom[c * HEADS + hh], __expf(v - m));
  }
}

// ---------------------------------------------------------------------------
// Edge pass 3: acc[col,:] += h[row,:] * alpha(head).  One wave32 per edge;
// lane handles 4 CONTIGUOUS channels (c0 = 4*lane) -> single global_load_b128
// of h[row] per lane (warp gather fully coalesced: 512B) and exactly ONE head
// per lane (hh = lane>>2) -> one exp + one divide per lane.
// ---------------------------------------------------------------------------
__global__ __launch_bounds__(256)
void edge_scatter(const int* __restrict__ ei, int E, int Nn,
                  const float* __restrict__ asrc,
                  const float* __restrict__ adst,
                  const unsigned* __restrict__ emax,
                  const float* __restrict__ denom,
                  const float* __restrict__ hmat,
                  float* __restrict__ acc) {
  int warp = (blockIdx.x * blockDim.x + threadIdx.x) >> 5;
  int lane = threadIdx.x & 31;
  const int EE = E + Nn;
  if (warp >= EE) return;
  int r, c;
  if (warp < E) { r = ei[warp]; c = ei[E + warp]; } else { r = c = warp - E; }

  const int hh = lane >> 2;              // head for this lane's 4 channels
  float v = lrelu(asrc[r * HEADS + hh] + adst[c * HEADS + hh]);
  float m = ordered_dec(emax[c * HEADS + hh]);
  float alpha = __expf(v - m) / (denom[c * HEADS + hh] + 1e-16f);

  const int c0 = lane << 2;
  float4 hv = *(const float4*)(hmat + (size_t)r * D128 + c0);
  float* dst = acc + (size_t)c * D128 + c0;
  atomicAdd(dst + 0, hv.x * alpha);
  atomicAdd(dst + 1, hv.y * alpha);
  atomicAdd(dst + 2, hv.z * alpha);
  atomicAdd(dst + 3, hv.w * alpha);
}

// ---------------------------------------------------------------------------
// Finalize layer 1: h1 = relu(acc + b1)   (float4 vectorized)
// ---------------------------------------------------------------------------
__global__ __launch_bounds__(256)
void finalize_relu(const float4* __restrict__ acc4, const float* __restrict__ bias,
                   float4* __restrict__ hout4, int Nn) {
  int i = blockIdx.x * blockDim.x + threadIdx.x;
  if (i >= Nn * (D128 / 4)) return;
  float4 a = acc4[i];
  float4 b = *(const float4*)(bias + ((i << 2) & (D128 - 1)));
  float4 o;
  o.x = fmaxf(a.x + b.x, 0.f);
  o.y = fmaxf(a.y + b.y, 0.f);
  o.z = fmaxf(a.z + b.z, 0.f);
  o.w = fmaxf(a.w + b.w, 0.f);
  hout4[i] = o;
}

// ---------------------------------------------------------------------------
// Finalize layer 2 + residual: out = x + h1 + relu(acc + b2)  (float4)
// ---------------------------------------------------------------------------
__global__ __launch_bounds__(256)
void final_sum(const float4* __restrict__ x4, const float4* __restrict__ h14,
               const float4* __restrict__ acc4, const float* __restrict__ bias,
               float4* __restrict__ out4, int Nn) {
  int i = blockIdx.x * blockDim.x + threadIdx.x;
  if (i >= Nn * (D128 / 4)) return;
  float4 a = acc4[i];
  float4 b = *(const float4*)(bias + ((i << 2) & (D128 - 1)));
  float4 xx = x4[i], h1 = h14[i];
  float4 o;
  o.x = xx.x + h1.x + fmaxf(a.x + b.x, 0.f);
  o.y = xx.y + h1.y + fmaxf(a.y + b.y, 0.f);
  o.z = xx.z + h1.z + fmaxf(a.z + b.z, 0.f);
  o.w = xx.w + h1.w + fmaxf(a.w + b.w, 0.f);
  out4[i] = o;
}

// ---------------------------------------------------------------------------

static inline int cdiv(int a, int b) { return (a + b - 1) / b; }

static void run_gat_layer(const float* xin, const int* ei, int E, int Nn,
                          const float* W, const float* av_src, const float* av_dst,
                          float* hbuf, float* asrc, float* adst,
                          unsigned* emax, float* denom, float* acc,
                          hipStream_t stream) {
  // 1. h = xin @ W  (WMMA)
  {
    int waves = cdiv(Nn, 16) * 8;
    gemm128_wmma_f32<<<cdiv(waves, 2), 64, 0, stream>>>(xin, W, hbuf, Nn);
  }
  // 2. per-node logits
  node_alpha<<<cdiv(Nn * HEADS, 256), 256, 0, stream>>>(hbuf, av_src, av_dst,
                                                        asrc, adst, Nn);
  // 3. reset accumulators
  init_bufs<<<cdiv(Nn * (D128 / 4), 256), 256, 0, stream>>>((float4*)acc, denom,
                                                            emax, Nn);
  // 4-6. edge passes (self-loops appended as edges [E, E+Nn))
  const int EE = E + Nn;
  edge_pass_max<<<cdiv(EE, 256), 256, 0, stream>>>(ei, E, Nn, asrc, adst, emax);
  edge_pass_denom<<<cdiv(EE, 256), 256, 0, stream>>>(ei, E, Nn, asrc, adst,
                                                     emax, denom);
  edge_scatter<<<cdiv(EE * 32, 256), 256, 0, stream>>>(ei, E, Nn, asrc, adst,
                                                       emax, denom, hbuf, acc);
}

extern "C" void kernel_launch(void* const* d_in, const int* in_sizes, int n_in,
                              void* d_out, int out_size, void* d_ws, size_t ws_size,
                              hipStream_t stream) {
  (void)n_in; (void)out_size; (void)ws_size;
  const float* x      = (const float*)d_in[0];
  const int*   ei     = (const int*)d_in[1];   // [2,E] int32
  // d_in[2] = pos  : dist embedding is dead code in the reference -> skipped
  const float* W1     = (const float*)d_in[3];
  const float* a_src1 = (const float*)d_in[4];
  const float* a_dst1 = (const float*)d_in[5];
  const float* b1     = (const float*)d_in[6];
  const float* W2     = (const float*)d_in[7];
  const float* a_src2 = (const float*)d_in[8];
  const float* a_dst2 = (const float*)d_in[9];
  const float* b2     = (const float*)d_in[10];
  float* out = (float*)d_out;

  const int Nn = in_sizes[0] / D128;
  const int E  = in_sizes[1] / 2;

  // --- workspace carve ---
  char* w = (char*)d_ws;
  float*    hbuf  = (float*)w;    w += (size_t)Nn * D128 * sizeof(float);
  float*    h1out = (float*)w;    w += (size_t)Nn * D128 * sizeof(float);
  float*    acc   = (float*)w;    w += (size_t)Nn * D128 * sizeof(float);
  float*    asrc  = (float*)w;    w += (size_t)Nn * HEADS * sizeof(float);
  float*    adst  = (float*)w;    w += (size_t)Nn * HEADS * sizeof(float);
  float*    denom = (float*)w;    w += (size_t)Nn * HEADS * sizeof(float);
  unsigned* emax  = (unsigned*)w;

  // ---- layer 1 ----
  run_gat_layer(x, ei, E, Nn, W1, a_src1, a_dst1,
                hbuf, asrc, adst, emax, denom, acc, stream);
  finalize_relu<<<cdiv(Nn * (D128 / 4), 256), 256, 0, stream>>>(
      (const float4*)acc, b1, (float4*)h1out, Nn);

  // ---- layer 2 (input = h1out) ----
  run_gat_layer(h1out, ei, E, Nn, W2, a_src2, a_dst2,
                hbuf, asrc, adst, emax, denom, acc, stream);

  // ---- out = x + h1 + relu(acc + b2) ----
  final_sum<<<cdiv(Nn * (D128 / 4), 256), 256, 0, stream>>>(
      (const float4*)x, (const float4*)h1out, (const float4*)acc, b2,
      (float4*)out, Nn);
}